// MiniMaceEmbedding_57750130262464
// MI455X (gfx1250) — compile-verified
//
#include <hip/hip_runtime.h>
#include <hip/hip_bf16.h>

// ---------------------------------------------------------------------------
// MiniMACE forward for MI455X (gfx1250).
//  - Dense per-atom linear algebra: V_WMMA_F32_16X16X4_F32, 5 waves/block,
//    weights staged in LDS (TDM tensor_load_to_lds when available, ds copy
//    fallback), K-loop unrolled x4.
//  - Edge scatter uses global_atomic_add_f32 (46MB density lives in 192MB L2).
//  - CG tensor product: sparse constant table, 1 thread per (atom, channel).
// ---------------------------------------------------------------------------

#define N_ATOMS 10000
#define N_EDGES 320000
#define DIM     128
#define NCH     16
#define MSG     16
#define RDIM    8
#define NCD     128     // MSG*RDIM
#define SH      9       // (LMAX+1)^2
#define NPATH   15
#define LAT_IN  288     // DIM + NCD + 2*NCH
#define CUTOFF  5.0f
#define PI_F    3.14159265358979f

#define GEMM_WAVES 5           // waves per block
#define GEMM_ROWS  (GEMM_WAVES * 16)   // 80 rows per block; 10000/80 = 125 exact
#define KMAX       288

typedef __attribute__((ext_vector_type(2))) float        v2f;
typedef __attribute__((ext_vector_type(8))) float        v8f;
typedef __attribute__((ext_vector_type(4))) unsigned int v4u;
typedef __attribute__((ext_vector_type(4))) int          v4i;
typedef __attribute__((ext_vector_type(8))) int          v8i;

// ---------------------------------------------------------------------------
// Strided fp32 WMMA GEMM: C[m,n] = act( sum_k A[m,k]*B[k,n] + bias[n] ) (+resid)
//   A[m,k] = A[m*lda + k*lda_k]
//   B[k,n] = B[k*ldb_k + n*ldb_n]   (staged into LDS once per block)
//   C[m,n] = C[m*ldc + n*ldc_n]
// Block = 160 threads = 5 waves; wave w owns rows [tm0+16w, tm0+16w+16).
// B tile (K x 16) for this block's N-tile is DMA'd into LDS with the Tensor
// Data Mover (D# built in-kernel), then every wave reads it via ds_load.
// WMMA lane layout per CDNA5 ISA 7.12.2:
//   A 16x4 : M = lane&15; VGPR j holds K = 2*(lane>>4)+j
//   B 4x16 : N = lane&15; VGPR j holds K = 2*(lane>>4)+j
//   C 16x16: M = v + 8*(lane>>4); N = lane&15
// ---------------------------------------------------------------------------
__global__ __launch_bounds__(GEMM_WAVES * 32) void gemm16_wmma(
    const float* __restrict__ A, int lda, int lda_k,
    const float* __restrict__ B, int ldb_k, int ldb_n,
    float*       __restrict__ C, int ldc, int ldc_n,
    const float* __restrict__ bias,
    const float* __restrict__ resid,
    int K, int act_silu,
    int d0, int bstride,      // B tile: d0 contiguous elems/row, row stride
    int lb_k, int lb_n)       // LDS strides so shB[k*lb_k + n*lb_n] = B[k,n]
{
  __shared__ float shB[KMAX * 16];

  const int tn   = blockIdx.x << 4;
  const int tm0  = blockIdx.y * GEMM_ROWS;
  const int tid  = threadIdx.x;
  const int wave = tid >> 5;
  const int lane = tid & 31;
  const int col  = lane & 15;
  const int hi   = lane >> 4;

  const float* Bt = B + (long)tn * ldb_n;   // B tile base: element (k=0, n=tn)
  const int nelem = K << 4;                 // K*16 elements in the tile

#if __has_builtin(__builtin_amdgcn_tensor_load_to_lds)
  // ---- Tensor Data Mover: async DMA B tile (2D, fp32) into LDS ----------
  if (wave == 0) {
    const unsigned long long ga = (unsigned long long)(uintptr_t)Bt;
    const unsigned int lds_off  = (unsigned int)(uintptr_t)&shB[0];
    const unsigned int ud0 = (unsigned int)d0;
    const unsigned int ud1 = (unsigned int)(nelem / d0);

    v4u g0;                                   // D# group 0 (128b)
    g0.x = 1u;                                // count=1, user descriptor
    g0.y = lds_off;                           // lds_addr
    g0.z = (unsigned int)(ga & 0xFFFFFFFFull);          // global_addr[31:0]
    g0.w = (unsigned int)((ga >> 32) & 0x1FFFFFFull)    // global_addr[56:32]
         | (2u << 30);                        // type = 2 ("image")

    v8i g1;                                   // D# group 1 (256b)
    g1[0] = (int)(2u << 16);                  // workgroup_mask=0, data_size=4B
    g1[1] = (int)(ud0 << 16);                 // tensor_dim0[15:0] in bits 31:16
    g1[2] = (int)((ud0 >> 16) | (ud1 << 16)); // tensor_dim0 hi | tensor_dim1 lo
    g1[3] = (int)((ud1 >> 16) | (ud0 << 16)); // tensor_dim1 hi | tile_dim0
    g1[4] = (int)(ud1 & 0xFFFFu);             // tile_dim1 ; tile_dim2 = 0
    g1[5] = bstride;                          // tensor_dim0_stride[31:0]
    g1[6] = 0;                                // stride hi | tensor_dim1_stride lo
    g1[7] = 0;

    const v4i z4 = {0, 0, 0, 0};
#if __clang_major__ >= 23
    const v8i z8 = {0, 0, 0, 0, 0, 0, 0, 0};
    __builtin_amdgcn_tensor_load_to_lds(g0, g1, z4, z4, z8, 0);
#else
    __builtin_amdgcn_tensor_load_to_lds(g0, g1, z4, z4, 0);
#endif
    __builtin_amdgcn_s_wait_tensorcnt(0);     // s_wait_tensorcnt 0
  }
  __syncthreads();
#else
  // ---- fallback: cooperative copy (same LDS layout as the TDM path) -----
  for (int t = tid; t < nelem; t += GEMM_WAVES * 32) {
    const int i1 = t / d0, i0 = t - i1 * d0;
    shB[t] = Bt[(long)i1 * bstride + i0];
  }
  __syncthreads();
#endif

  const int myTm = tm0 + (wave << 4);
  const float* Arow = A + (long)(myTm + col) * lda;

  v8f acc = {};
  for (int k0 = 0; k0 < K; k0 += 16) {
    if (k0 + 16 < K)
      __builtin_prefetch(&Arow[(long)(k0 + 16) * lda_k], 0, 1);  // global_prefetch_b8
#pragma unroll
    for (int u = 0; u < 4; ++u) {
      const int ka = k0 + (u << 2) + (hi << 1);
      v2f a, b;
      a.x = Arow[(long)ka       * lda_k];
      a.y = Arow[(long)(ka + 1) * lda_k];
      b.x = shB[ka       * lb_k + col * lb_n];   // ds_load_b32
      b.y = shB[(ka + 1) * lb_k + col * lb_n];
      acc = __builtin_amdgcn_wmma_f32_16x16x4_f32(false, a, false, b,
                                                  (short)0, acc, false, false);
    }
  }

  const int cn = tn + col;
  const float bs = bias ? bias[cn] : 0.0f;
#pragma unroll
  for (int v = 0; v < 8; ++v) {
    const int  rm  = myTm + v + (hi << 3);
    const long idx = (long)rm * ldc + (long)cn * ldc_n;
    float x = acc[v] + bs;
    if (act_silu) x = x * __builtin_amdgcn_rcpf(1.0f + __expf(-x));  // SiLU
    if (resid)    x += resid[idx];
    C[idx] = x;
  }
}

// ---------------------------------------------------------------------------
// Edge geometry: Bessel radial basis (x switch) and real spherical harmonics.
// ---------------------------------------------------------------------------
__global__ void edge_geom(const float* __restrict__ dist,
                          const float* __restrict__ vec,
                          const float* __restrict__ sw,
                          float* __restrict__ radial,
                          float* __restrict__ Y)
{
  int e = blockIdx.x * blockDim.x + threadIdx.x;
  if (e >= N_EDGES) return;
  const float r = dist[e];
  const float invr = __builtin_amdgcn_rcpf(r);
  const float pref = sqrtf(2.0f / CUTOFF) * invr * sw[e];
#pragma unroll
  for (int n = 0; n < RDIM; ++n)
    radial[e * RDIM + n] = pref * __sinf((float)(n + 1) * PI_F * r / CUTOFF);

  const float x = vec[3 * e + 0] * invr;
  const float y = vec[3 * e + 1] * invr;
  const float z = vec[3 * e + 2] * invr;
  const float s3 = 1.73205081f, s5 = 2.23606798f, s15 = 3.87298335f;
  float* yo = Y + (long)e * SH;
  yo[0] = 1.0f;
  yo[1] = s3 * y;  yo[2] = s3 * z;  yo[3] = s3 * x;
  yo[4] = s15 * x * y;
  yo[5] = s15 * y * z;
  yo[6] = 0.5f * s5 * (3.0f * z * z - 1.0f);
  yo[7] = s15 * x * z;
  yo[8] = 0.5f * s15 * (x * x - y * y);
}

// ---------------------------------------------------------------------------
// Atom embedding: xi = emb_W[species] + emb_b
// ---------------------------------------------------------------------------
__global__ void embed_kernel(const int* __restrict__ species,
                             const float* __restrict__ W,
                             const float* __restrict__ b,
                             float* __restrict__ xi)
{
  const int n = blockIdx.x, d = threadIdx.x;
  xi[(long)n * DIM + d] = W[(long)species[n] * DIM + d] + b[d];
}

// ---------------------------------------------------------------------------
// Layer-0 edge scatter: density[src] += (mi[dst] (x) radial) (x) Y
// One block per edge, one thread per channel c (0..127).
// ---------------------------------------------------------------------------
__global__ void edge_density0(const int* __restrict__ src,
                              const int* __restrict__ dst,
                              const float* __restrict__ radial,
                              const float* __restrict__ Y,
                              const float* __restrict__ mi,
                              float* __restrict__ density)
{
  const int e = blockIdx.x, c = threadIdx.x;
  const int s = src[e], d = dst[e];
  const float xc = mi[(long)d * MSG + (c >> 3)] * radial[(long)e * RDIM + (c & 7)];
  const float* y = Y + (long)e * SH;
  float* drow = density + (long)s * NCD * SH + (long)c * SH;
#pragma unroll
  for (int m = 0; m < SH; ++m)
    atomicAdd(drow + m, xc * y[m]);
}

// Layer-1 edge scatter: density[src] += (mi[dst] (x) radial) * rhoi[dst]
__global__ void edge_density1(const int* __restrict__ src,
                              const int* __restrict__ dst,
                              const float* __restrict__ radial,
                              const float* __restrict__ rhoi,
                              const float* __restrict__ mi,
                              float* __restrict__ density)
{
  const int e = blockIdx.x, c = threadIdx.x;
  const int s = src[e], d = dst[e];
  const float xc = mi[(long)d * MSG + (c >> 3)] * radial[(long)e * RDIM + (c & 7)];
  const float* rrow = rhoi + (long)d * NCD * SH + (long)c * SH;
  float* drow = density + (long)s * NCD * SH + (long)c * SH;
#pragma unroll
  for (int m = 0; m < SH; ++m)
    atomicAdd(drow + m, xc * rrow[m]);
}

// ---------------------------------------------------------------------------
// Sparse real-Clebsch-Gordan table (generated offline from _real_cg()).
// Absolute SH indices: l=0 -> 0 ; l=1 -> 1..3 (y,z,x) ; l=2 -> 4..8.
// Entry contributes: L[c] += v * w[p] * V[a] * H[b]
// ---------------------------------------------------------------------------
struct CGEntry { unsigned char a, b, c, p; float v; };

__constant__ CGEntry CG_TAB[] = {
  // p0 (0,0,0)
  {0,0,0, 0, 1.0f},
  // p1 (0,1,1)
  {0,1,1, 1, 1.0f},{0,2,2, 1, 1.0f},{0,3,3, 1, 1.0f},
  // p2 (0,2,2)
  {0,4,4, 2, 1.0f},{0,5,5, 2, 1.0f},{0,6,6, 2, 1.0f},{0,7,7, 2, 1.0f},{0,8,8, 2, 1.0f},
  // p3 (1,0,1)
  {1,0,1, 3, 1.0f},{2,0,2, 3, 1.0f},{3,0,3, 3, 1.0f},
  // p4 (1,1,0)
  {1,1,0, 4, 0.57735027f},{2,2,0, 4, 0.57735027f},{3,3,0, 4, 0.57735027f},
  // p5 (1,1,1) antisymmetric (cross-product structure)
  {2,3,1, 5, 0.70710678f},{3,2,1, 5,-0.70710678f},
  {3,1,2, 5, 0.70710678f},{1,3,2, 5,-0.70710678f},
  {1,2,3, 5, 0.70710678f},{2,1,3, 5,-0.70710678f},
  // p6 (1,1,2) symmetric traceless
  {3,1,4, 6, 0.70710678f},{1,3,4, 6, 0.70710678f},
  {1,2,5, 6, 0.70710678f},{2,1,5, 6, 0.70710678f},
  {2,2,6, 6, 0.81649658f},{1,1,6, 6,-0.40824829f},{3,3,6, 6,-0.40824829f},
  {3,2,7, 6, 0.70710678f},{2,3,7, 6, 0.70710678f},
  {3,3,8, 6, 0.70710678f},{1,1,8, 6,-0.70710678f},
  // p7 (1,2,1)
  {1,4,3, 7, 0.54772256f},{3,4,1, 7, 0.54772256f},
  {2,5,1, 7, 0.54772256f},{1,5,2, 7, 0.54772256f},
  {1,6,1, 7,-0.31622777f},{2,6,2, 7, 0.63245553f},{3,6,3, 7,-0.31622777f},
  {2,7,3, 7, 0.54772256f},{3,7,2, 7, 0.54772256f},
  {3,8,3, 7, 0.54772256f},{1,8,1, 7,-0.54772256f},
  // p8 (1,2,2)
  {1,5,8, 8, 0.54772256f},{1,7,6, 8, 0.54772256f},{1,4,7, 8,-0.54772256f},{1,8,5, 8,-0.54772256f},
  {2,4,4, 8, 0.54772256f},{2,8,8, 8,-0.54772256f},{2,5,7, 8, 0.54772256f},{2,7,5, 8,-0.54772256f},
  {3,5,4, 8,-0.54772256f},{3,4,5, 8, 0.54772256f},{3,7,8, 8, 0.54772256f},{3,8,7, 8,-0.54772256f},
  // p9 (2,0,2)
  {4,0,4, 9, 1.0f},{5,0,5, 9, 1.0f},{6,0,6, 9, 1.0f},{7,0,7, 9, 1.0f},{8,0,8, 9, 1.0f},
  // p10 (2,1,1)
  {4,1,3,10, 0.54772256f},{4,3,1,10, 0.54772256f},
  {5,2,1,10, 0.54772256f},{5,1,2,10, 0.54772256f},
  {6,1,1,10,-0.31622777f},{6,2,2,10, 0.63245553f},{6,3,3,10,-0.31622777f},
  {7,2,3,10, 0.54772256f},{7,3,2,10, 0.54772256f},
  {8,3,3,10, 0.54772256f},{8,1,1,10,-0.54772256f},
  // p11 (2,1,2)
  {5,1,8,11, 0.54772256f},{7,1,6,11, 0.54772256f},{4,1,7,11,-0.54772256f},{8,1,5,11,-0.54772256f},
  {4,2,4,11, 0.54772256f},{8,2,8,11,-0.54772256f},{5,2,7,11, 0.54772256f},{7,2,5,11,-0.54772256f},
  {5,3,4,11,-0.54772256f},{4,3,5,11, 0.54772256f},{7,3,8,11, 0.54772256f},{8,3,7,11,-0.54772256f},
  // p12 (2,2,0)
  {4,4,0,12, 0.44721360f},{5,5,0,12, 0.44721360f},{6,6,0,12, 0.44721360f},
  {7,7,0,12, 0.44721360f},{8,8,0,12, 0.44721360f},
  // p13 (2,2,1) antisymmetric
  {4,5,3,13, 0.54772256f},{5,4,3,13,-0.54772256f},
  {5,8,1,13, 0.54772256f},{8,5,1,13,-0.54772256f},
  {6,5,2,13, 0.54772256f},{5,6,2,13,-0.54772256f},
  {7,8,2,13, 0.54772256f},{8,7,2,13,-0.54772256f},
  {4,7,1,13, 0.54772256f},{7,4,1,13,-0.54772256f},
  // p14 (2,2,2) symmetric
  {4,6,4,14,-0.53452248f},{6,4,4,14,-0.53452248f},{5,7,4,14, 0.46291005f},{7,5,4,14, 0.46291005f},
  {5,6,5,14, 0.26726124f},{6,5,5,14, 0.26726124f},{4,7,5,14, 0.46291005f},{7,4,5,14, 0.46291005f},
  {6,6,6,14, 0.53452248f},{4,4,6,14,-0.53452248f},{8,8,6,14,-0.53452248f},
  {5,5,6,14, 0.26726124f},{7,7,6,14, 0.26726124f},
  {7,6,7,14, 0.26726124f},{6,7,7,14, 0.26726124f},
  {8,6,8,14,-0.53452248f},{6,8,8,14,-0.53452248f},
};
constexpr int CG_N = (int)(sizeof(CG_TAB) / sizeof(CGEntry));

// ---------------------------------------------------------------------------
// filtered_tp: Li[n,i,:] = sum_paths CG . V[n,i,:] . H[n,i,:] * w[p,i]
// Then Vi += Li. One thread per (atom, channel).
// ---------------------------------------------------------------------------
__global__ void ftp_kernel(float* __restrict__ Vi,
                           const float* __restrict__ Hi,
                           const float* __restrict__ tpw,   // [NPATH, NCH]
                           float* __restrict__ Li)
{
  const int idx = blockIdx.x * blockDim.x + threadIdx.x;
  if (idx >= N_ATOMS * NCH) return;
  const int n = idx / NCH, i = idx % NCH;

  float V[SH], H[SH], L[SH], w[NPATH];
  const long base = (long)n * NCH * SH + (long)i * SH;
#pragma unroll
  for (int m = 0; m < SH; ++m) { V[m] = Vi[base + m]; H[m] = Hi[base + m]; L[m] = 0.0f; }
#pragma unroll
  for (int p = 0; p < NPATH; ++p) w[p] = tpw[p * NCH + i];

  for (int t = 0; t < CG_N; ++t) {
    const CGEntry e = CG_TAB[t];
    L[e.c] += e.v * w[e.p] * V[e.a] * H[e.b];
  }
#pragma unroll
  for (int m = 0; m < SH; ++m) {
    Li[base + m] = L[m];
    Vi[base + m] = V[m] + L[m];
  }
}

// ---------------------------------------------------------------------------
// Pack latent input: h = [xi(128) | density[...,0](128) | Li0[...,0](16) | Li1[...,0](16)]
// ---------------------------------------------------------------------------
__global__ void pack_h(const float* __restrict__ xi,
                       const float* __restrict__ density,
                       const float* __restrict__ Li0,
                       const float* __restrict__ Li1,
                       float* __restrict__ h)
{
  const int n = blockIdx.x, c = threadIdx.x;
  float v;
  if (c < DIM)            v = xi[(long)n * DIM + c];
  else if (c < DIM + NCD) v = density[(long)n * NCD * SH + (long)(c - DIM) * SH];
  else if (c < DIM + NCD + NCH)
                          v = Li0[(long)n * NCH * SH + (long)(c - DIM - NCD) * SH];
  else                    v = Li1[(long)n * NCH * SH + (long)(c - DIM - NCD - NCH) * SH];
  h[(long)n * LAT_IN + c] = v;
}

// ---------------------------------------------------------------------------
// Host-side orchestration
// ---------------------------------------------------------------------------
static inline void launch_gemm(hipStream_t s,
                               const float* A, int lda, int lda_k,
                               const float* B, int ldb_k, int ldb_n,
                               float* C, int ldc, int ldc_n,
                               const float* bias, const float* resid,
                               int M, int N, int K, int silu)
{
  // B-tile staging geometry: which axis of B is contiguous in memory.
  int d0, bstride, lb_k, lb_n;
  if (ldb_n == 1) { d0 = 16; bstride = ldb_k; lb_k = 16; lb_n = 1; }  // n contiguous
  else            { d0 = K;  bstride = ldb_n; lb_k = 1;  lb_n = K; }  // k contiguous
  dim3 grid(N / 16, M / GEMM_ROWS);
  gemm16_wmma<<<grid, GEMM_WAVES * 32, 0, s>>>(A, lda, lda_k, B, ldb_k, ldb_n,
                                               C, ldc, ldc_n, bias, resid,
                                               K, silu, d0, bstride, lb_k, lb_n);
}

extern "C" void kernel_launch(void* const* d_in, const int* in_sizes, int n_in,
                              void* d_out, int out_size, void* d_ws, size_t ws_size,
                              hipStream_t stream)
{
  // ---- inputs (setup_inputs dict order; params dict flattened in order) ----
  const int*   species = (const int*)  d_in[0];
  const int*   esrc    = (const int*)  d_in[1];
  const int*   edst    = (const int*)  d_in[2];
  const float* dist    = (const float*)d_in[3];
  const float* vec     = (const float*)d_in[4];
  const float* sw      = (const float*)d_in[5];
  const float* emb_W   = (const float*)d_in[6];   // (64,128)
  const float* emb_b   = (const float*)d_in[7];   // (128,)
  const float* vi_W    = (const float*)d_in[8];   // (3,16,128)
  int pi = 9;
  const float *lin_W[2], *lin_b[2], *rho_W[2] = {nullptr, nullptr};
  const float *mix_W[2][2], *tp_w[2][2];
  const float *lat_W0[2], *lat_b0[2], *lat_W1[2], *lat_b1[2];
  for (int l = 0; l < 2; ++l) {
    lin_W[l] = (const float*)d_in[pi++];          // (128,16)
    lin_b[l] = (const float*)d_in[pi++];          // (16,)
    if (l > 0) rho_W[l] = (const float*)d_in[pi++]; // (3,128,16)
    for (int i = 0; i < 2; ++i) {
      mix_W[l][i] = (const float*)d_in[pi++];     // (16,128)
      tp_w[l][i]  = (const float*)d_in[pi++];     // (15,16)
    }
    lat_W0[l] = (const float*)d_in[pi++];         // (288,128)
    lat_b0[l] = (const float*)d_in[pi++];         // (128,)
    lat_W1[l] = (const float*)d_in[pi++];         // (128,128)
    lat_b1[l] = (const float*)d_in[pi++];         // (128,)
  }

  // ---- workspace carve-up (floats) ----
  float* ws      = (float*)d_ws;
  float* radial  = ws; ws += (size_t)N_EDGES * RDIM;
  float* Y       = ws; ws += (size_t)N_EDGES * SH;
  float* xi      = ws; ws += (size_t)N_ATOMS * DIM;
  float* mi      = ws; ws += (size_t)N_ATOMS * MSG;
  float* density = ws; ws += (size_t)N_ATOMS * NCD * SH;
  float* Vi      = ws; ws += (size_t)N_ATOMS * NCH * SH;
  float* rhoi    = ws; ws += (size_t)N_ATOMS * NCD * SH;
  float* Hi      = ws; ws += (size_t)N_ATOMS * NCH * SH;
  float* Li0     = ws; ws += (size_t)N_ATOMS * NCH * SH;
  float* Li1     = ws; ws += (size_t)N_ATOMS * NCH * SH;
  float* h       = ws; ws += (size_t)N_ATOMS * LAT_IN;
  float* hmid    = ws; ws += (size_t)N_ATOMS * DIM;

  // ---- edge geometry + embedding ----
  edge_geom<<<(N_EDGES + 255) / 256, 256, 0, stream>>>(dist, vec, sw, radial, Y);
  embed_kernel<<<N_ATOMS, DIM, 0, stream>>>(species, emb_W, emb_b, xi);

  for (int l = 0; l < 2; ++l) {
    // mi = xi @ lin_W + lin_b   (10000x128)@(128x16)
    launch_gemm(stream, xi, DIM, 1, lin_W[l], MSG, 1, mi, MSG, 1,
                lin_b[l], nullptr, N_ATOMS, MSG, DIM, 0);

    if (l == 0) {
      hipMemsetAsync(density, 0, (size_t)N_ATOMS * NCD * SH * sizeof(float), stream);
      edge_density0<<<N_EDGES, NCD, 0, stream>>>(esrc, edst, radial, Y, mi, density);
      // Vi = cm_e3(density, vi_W): per SH component j, (10000x128)@(128x16)
      for (int ll = 0; ll <= 2; ++ll)
        for (int mm = 0; mm < 2 * ll + 1; ++mm) {
          const int j = ll * ll + mm;
          launch_gemm(stream,
                      density + j, NCD * SH, SH,              // A[n,c] strided by SH
                      vi_W + (size_t)ll * NCH * NCD, 1, NCD,  // B[c,o] = W[o,c]
                      Vi + j, NCH * SH, SH,
                      nullptr, nullptr, N_ATOMS, NCH, NCD, 0);
        }
    } else {
      // rhoi = cm_e3(Vi, rho_W): per j, (10000x16)@(16x128)
      for (int ll = 0; ll <= 2; ++ll)
        for (int mm = 0; mm < 2 * ll + 1; ++mm) {
          const int j = ll * ll + mm;
          launch_gemm(stream,
                      Vi + j, NCH * SH, SH,
                      rho_W[l] + (size_t)ll * NCD * NCH, 1, NCH,  // B[c,o] = W[o,c]
                      rhoi + j, NCD * SH, SH,
                      nullptr, nullptr, N_ATOMS, NCD, NCH, 0);
        }
      edge_density1<<<N_EDGES, NCD, 0, stream>>>(esrc, edst, radial, rhoi, mi, density);
    }

    // Tensor products (sequential in i: Vi updated between them)
    for (int i = 0; i < 2; ++i) {
      for (int j = 0; j < SH; ++j)
        launch_gemm(stream,
                    density + j, NCD * SH, SH,
                    mix_W[l][i], 1, NCD,                  // B[c,o] = mixW[o,c]
                    Hi + j, NCH * SH, SH,
                    nullptr, nullptr, N_ATOMS, NCH, NCD, 0);
      ftp_kernel<<<(N_ATOMS * NCH + 255) / 256, 256, 0, stream>>>(
          Vi, Hi, tp_w[l][i], (i == 0) ? Li0 : Li1);
    }

    // Latent MLP with residual
    pack_h<<<N_ATOMS, LAT_IN, 0, stream>>>(xi, density, Li0, Li1, h);
    launch_gemm(stream, h, LAT_IN, 1, lat_W0[l], DIM, 1, hmid, DIM, 1,
                lat_b0[l], nullptr, N_ATOMS, DIM, LAT_IN, 1 /*silu*/);
    launch_gemm(stream, hmid, DIM, 1, lat_W1[l], DIM, 1, xi, DIM, 1,
                lat_b1[l], xi /*residual*/, N_ATOMS, DIM, DIM, 0);
  }

  // ---- outputs: (xi, Vi) flattened ----
  float* out = (float*)d_out;
  hipMemcpyAsync(out, xi, (size_t)N_ATOMS * DIM * sizeof(float),
                 hipMemcpyDeviceToDevice, stream);
  hipMemcpyAsync(out + (size_t)N_ATOMS * DIM, Vi,
                 (size_t)N_ATOMS * NCH * SH * sizeof(float),
                 hipMemcpyDeviceToDevice, stream);
}